// SequenceLoss_23227183137436
// MI455X (gfx1250) — compile-verified
//
#include <hip/hip_runtime.h>
#include <hip/hip_bf16.h>

typedef __attribute__((ext_vector_type(2))) float v2f;
typedef __attribute__((ext_vector_type(8))) float v8f;

#define NB    32
#define NL    200
#define ND    64
#define NNEG  100
#define NPOS  (NB * NL)          // 6400 positions
#define WPB   8                  // waves per block (256 threads)

// --- zero the two f32 accumulators in workspace (graph-capture safe) ---
__global__ void bpr_init_ws(float* ws) {
    ws[0] = 0.0f;   // sum of masked loss
    ws[1] = 0.0f;   // sum of mask weights (denominator)
}

// --- main kernel: one wave32 per (b,l) position ---
__global__ __launch_bounds__(256) void bpr_wmma_kernel(
    const float* __restrict__ seq_embs,   // (B,L,D) f32
    const int*   __restrict__ target_seq, // (B,L)
    const float* __restrict__ mask,       // (B,L) f32
    const int*   __restrict__ neg_items,  // (B,L,N)
    const float* __restrict__ emb_table,  // (VOCAB,D) f32
    float*       __restrict__ ws)
{
    const int lane = threadIdx.x & 31;
    const int wave = threadIdx.x >> 5;
    const int p    = blockIdx.x * WPB + wave;   // wave-uniform
    if (p >= NPOS) return;                      // uniform per wave; EXEC stays all-ones

    const float* seqp = seq_embs + (size_t)p * ND;
    const int    base = (lane >> 4) << 1;       // lanes 0-15 -> cols k0+0,1 ; lanes 16-31 -> k0+2,3

    // Preload the seq vector chunks this lane needs for the B operand
    // (B[k,n] = seq[k] replicated across all 16 columns).
    v2f sb[16];
#pragma unroll
    for (int c = 0; c < 16; ++c)
        sb[c] = *(const v2f*)(seqp + c * 4 + base);

    // ---- positive score: exact f32 wave-wide dot product ----
    const int    tgt = target_seq[p];
    const float* pep = emb_table + (size_t)tgt * ND;
    v2f s2 = *(const v2f*)(seqp + 2 * lane);
    v2f e2 = *(const v2f*)(pep  + 2 * lane);
    float part = s2.x * e2.x + s2.y * e2.y;
#pragma unroll
    for (int off = 16; off >= 1; off >>= 1)
        part += __shfl_xor(part, off, 32);
    const float pos = part;                     // broadcast to all lanes

    // ---- negatives: 7 groups of 16 rows via V_WMMA_F32_16X16X4_F32 ----
    const int m = lane & 15;                    // A-matrix row this lane loads
    float lsum = 0.0f;

#pragma unroll 1
    for (int g = 0; g < 7; ++g) {
        int nidx = g * 16 + m;
        int item = neg_items[(size_t)p * NNEG + (nidx < NNEG ? nidx : 0)];
        const float* rp = emb_table + (size_t)item * ND;

        // prefetch next group's gathered row (global_prefetch_b8)
        if (g < 6) {
            int nn  = (g + 1) * 16 + m;
            int it2 = neg_items[(size_t)p * NNEG + (nn < NNEG ? nn : 0)];
            __builtin_prefetch(emb_table + (size_t)it2 * ND, 0, 1);
        }

        v8f acc = {0.f, 0.f, 0.f, 0.f, 0.f, 0.f, 0.f, 0.f};
#pragma unroll
        for (int c = 0; c < 16; ++c) {
            v2f a = *(const v2f*)(rp + c * 4 + base);
            // D = A(16x4 f32) * B(4x16 f32) + C   — exact f32 accumulate
            acc = __builtin_amdgcn_wmma_f32_16x16x4_f32(
                /*neg_a=*/false, a, /*neg_b=*/false, sb[c],
                /*c_mod=*/(short)0, acc, /*reuse_a=*/false, /*reuse_b=*/false);
        }

        // lanes 0-15 hold negatives g*16+0..7 ; lanes 16-31 hold g*16+8..15
        const int mbase = g * 16 + ((lane >> 4) << 3);
#pragma unroll
        for (int r = 0; r < 8; ++r) {
            if (mbase + r < NNEG) {
                float d  = pos - acc[r];
                float sg = 1.0f / (1.0f + __expf(-d));
                lsum += -__logf(sg + 1e-8f);
            }
        }
    }

    // wave reduce (every valid negative was counted 16x, once per D column)
#pragma unroll
    for (int off = 16; off >= 1; off >>= 1)
        lsum += __shfl_xor(lsum, off, 32);

    if (lane == 0) {
        float w = mask[p];
        atomicAdd(&ws[0], (lsum * (1.0f / 16.0f)) * w);
        atomicAdd(&ws[1], w * (float)NNEG);
    }
}

__global__ void bpr_finalize(const float* __restrict__ ws, float* __restrict__ out) {
    out[0] = ws[0] / ws[1];
}

extern "C" void kernel_launch(void* const* d_in, const int* in_sizes, int n_in,
                              void* d_out, int out_size, void* d_ws, size_t ws_size,
                              hipStream_t stream) {
    const float* seq_embs   = (const float*)d_in[0];
    const int*   target_seq = (const int*)  d_in[1];
    const float* mask       = (const float*)d_in[2];
    const int*   neg_items  = (const int*)  d_in[3];
    const float* emb_table  = (const float*)d_in[4];
    float* out = (float*)d_out;
    float* ws  = (float*)d_ws;

    bpr_init_ws<<<1, 1, 0, stream>>>(ws);
    bpr_wmma_kernel<<<NPOS / WPB, 256, 0, stream>>>(
        seq_embs, target_seq, mask, neg_items, emb_table, ws);
    bpr_finalize<<<1, 1, 0, stream>>>(ws, out);
}